// LengthRegulator_9156870275497
// MI455X (gfx1250) — compile-verified
//
#include <hip/hip_runtime.h>
#include <hip/hip_bf16.h>
#include <stdint.h>

#define B_ 32
#define S_ 1024
#define D_ 512

// ---- probe: gfx1250 async global<->LDS builtins (verified present) ----
#if defined(__has_builtin)
#if __has_builtin(__builtin_amdgcn_global_load_async_to_lds_b128) && \
    __has_builtin(__builtin_amdgcn_global_store_async_from_lds_b128) && \
    __has_builtin(__builtin_amdgcn_s_wait_asynccnt)
#define USE_ASYNC_LDS 1
#else
#define USE_ASYNC_LDS 0
#endif
#else
#define USE_ASYNC_LDS 0
#endif

// Builtin signature (from hipcc diagnostic): pointers to 16-byte int vectors,
// global (AS1) first, LDS (AS3) second, then imm offset + imm cpol.
typedef int v4i __attribute__((vector_size(16)));
typedef __attribute__((address_space(1))) v4i* gv4i_p;
typedef __attribute__((address_space(3))) v4i* lv4i_p;

// cum layout: B rows of (S_+1) ints; crow[0] = 0, crow[s+1] = inclusive sum.
#define CUMW (S_ + 1)

// ---------------------------------------------------------------------------
// Scatter: one 128-thread block per (b,s); each lane moves 16B (B128), so one
// block moves exactly one 2KB feature row. Row is parked in LDS via async
// load, then fanned out to [crow[s], crow[s+1]) output rows via async stores
// (no VGPR round-trip per repeat). Falls back to float4 register copy.
// ---------------------------------------------------------------------------
__global__ void __launch_bounds__(128)
lr_scatter_kernel(const float* __restrict__ x, const int* __restrict__ cum,
                  float* __restrict__ out, int max_len) {
  const int bs = blockIdx.x;         // 0 .. B*S-1
  const int b = bs >> 10;            // S_ == 1024
  const int s = bs & (S_ - 1);
  const int* crow = cum + b * CUMW;
  const int start = crow[s];
  const int end = crow[s + 1];

  const float* src = x + (size_t)bs * D_;
  float* dstb = out + (size_t)b * max_len * D_;
  const int tid = threadIdx.x;

#if USE_ASYNC_LDS
  __shared__ __align__(16) float buf[D_];
  gv4i_p g_src = (gv4i_p)(src + tid * 4);
  lv4i_p l_buf = (lv4i_p)(&buf[tid * 4]);
  __builtin_amdgcn_global_load_async_to_lds_b128(g_src, l_buf, 0, 0);
  __builtin_amdgcn_s_wait_asynccnt(0);  // per-wave: our LDS bytes are resident
  for (int p = start; p < end; ++p) {
    gv4i_p g_dst = (gv4i_p)(dstb + (size_t)p * D_ + tid * 4);
    __builtin_amdgcn_global_store_async_from_lds_b128(g_dst, l_buf, 0, 0);
  }
  // S_ENDPGM performs an implicit wait-idle; async stores retire before exit.
#else
  const float4 v = ((const float4*)src)[tid];
  for (int p = start; p < end; ++p) {
    ((float4*)(dstb + (size_t)p * D_))[tid] = v;
  }
#endif
}

// ---------------------------------------------------------------------------
// Scan: per-batch inclusive scan of clip(dur,1,.) -> cum[B][S+1], totals[B].
// One block per batch; 256 threads x 4 elements each; Hillis-Steele on 256.
// ---------------------------------------------------------------------------
__global__ void __launch_bounds__(256)
lr_scan_kernel(const int* __restrict__ dur, int* __restrict__ cum,
               int* __restrict__ totals) {
  const int b = blockIdx.x;
  const int t = threadIdx.x;
  const int* drow = dur + b * S_;
  const int i0 = t * 4;

  int v0 = drow[i0 + 0]; v0 = v0 < 1 ? 1 : v0;
  int v1 = drow[i0 + 1]; v1 = v1 < 1 ? 1 : v1;
  int v2 = drow[i0 + 2]; v2 = v2 < 1 ? 1 : v2;
  int v3 = drow[i0 + 3]; v3 = v3 < 1 ? 1 : v3;

  const int s0 = v0, s1 = s0 + v1, s2 = s1 + v2, s3 = s2 + v3;

  __shared__ int part[256];
  part[t] = s3;
  __syncthreads();
  for (int off = 1; off < 256; off <<= 1) {
    int add = (t >= off) ? part[t - off] : 0;
    __syncthreads();
    part[t] += add;
    __syncthreads();
  }
  const int base = part[t] - s3;  // exclusive prefix of this thread's chunk

  int* crow = cum + b * CUMW;
  if (t == 0) crow[0] = 0;
  crow[i0 + 1] = base + s0;
  crow[i0 + 2] = base + s1;
  crow[i0 + 3] = base + s2;
  crow[i0 + 4] = base + s3;
  if (t == 255) totals[b] = part[255];
}

// ---------------------------------------------------------------------------
// Zero tail: rows p in [totals[b], max_len) only.
// Blocks below the tail exit after one scalar compare.
// ---------------------------------------------------------------------------
__global__ void __launch_bounds__(128)
lr_zerotail_kernel(const int* __restrict__ totals, float* __restrict__ out,
                   int max_len) {
  const int b = blockIdx.x;
  const int p = blockIdx.y;
  if (p < totals[b]) return;
  const float4 z = make_float4(0.f, 0.f, 0.f, 0.f);
  ((float4*)(out + ((size_t)b * max_len + p) * D_))[threadIdx.x] = z;
}

// ---------------------------------------------------------------------------
extern "C" void kernel_launch(void* const* d_in, const int* in_sizes, int n_in,
                              void* d_out, int out_size, void* d_ws,
                              size_t ws_size, hipStream_t stream) {
  const float* x = (const float*)d_in[0];
  const int* dur = (const int*)d_in[1];
  float* out = (float*)d_out;

  const int max_len = out_size / (B_ * D_);  // out is (B, max_len, D)

  int* cum = (int*)d_ws;            // B*(S+1) int32
  int* totals = cum + B_ * CUMW;    // B int32

  lr_scan_kernel<<<B_, 256, 0, stream>>>(dur, cum, totals);
  lr_zerotail_kernel<<<dim3(B_, max_len), 128, 0, stream>>>(totals, out,
                                                            max_len);
  lr_scatter_kernel<<<B_ * S_, 128, 0, stream>>>(x, cum, out, max_len);
}